// TernaryLoss_86672440033752
// MI455X (gfx1250) — compile-verified
//
#include <hip/hip_runtime.h>
#include <cstdint>
#include <cstddef>

// ---------------- problem constants ----------------
#define BATCH 16
#define H_IMG 256
#define W_IMG 448
// Owner-pixel domain (antisymmetry half-sum): q in [-2, H-2] x [-2, W+1].
// Tiles of 64x16 with origin shifted by -2: 8 x-tiles, 17 y-tiles.
#define TILE_W 64
#define TILE_H 16
#define HALO_L 4            // need 3 (dx=-3); +1 keeps float4 windows 16B aligned
#define HALO_R 3            // dx=+3
#define HALO_B 3            // dy=+3 (no top halo: half-set has dy>=0)
#define G_H    (TILE_H + HALO_B)            // 19 rows
#define G_COLS (TILE_W + HALO_L + HALO_R)   // 71 valid cols
#define G_W    72                           // padded row stride (288B, 16B multiple)
#define TILES_X 8
#define TILES_Y 17
#define NBLOCKS (TILES_X * TILES_Y * BATCH) // 2176

typedef __attribute__((address_space(3))) float lds_f32_t;

// gfx1250 async global->LDS copy (ASYNCcnt-tracked), CDNA5 ISA.
__device__ __forceinline__ void async_load_f32(const float* g, float* l) {
    unsigned lds_off = (unsigned)(unsigned long long)(lds_f32_t*)l; // LDS byte offset
    unsigned long long ga = (unsigned long long)g;                  // global address
    asm volatile("global_load_async_to_lds_b32 %0, %1, off"
                 :: "v"(lds_off), "v"(ga)
                 : "memory");
}
__device__ __forceinline__ void wait_async0() {
    asm volatile("s_wait_asynccnt 0" ::: "memory");
}

__global__ __launch_bounds__(256)
void census_loss_kernel(const float* __restrict__ pred,
                        const float* __restrict__ target,
                        float* __restrict__ partials)
{
    __shared__ alignas(16) float s_raw[6][G_H][G_W]; // r,g,b (pred) ; r,g,b (target)
    __shared__ alignas(16) float s_gp[G_H][G_W];     // gray(pred),  0 outside image
    __shared__ alignas(16) float s_gt[G_H][G_W];     // gray(target),0 outside image
    __shared__ float s_red[8];

    const int tid = threadIdx.x;
    const int qx0 = blockIdx.x * TILE_W - 2;  // owner-domain tile origin
    const int qy0 = blockIdx.y * TILE_H - 2;
    const int b   = blockIdx.z;

    const size_t plane = (size_t)H_IMG * W_IMG;
    const float* pbase = pred   + (size_t)b * 3 * plane;
    const float* tbase = target + (size_t)b * 3 * plane;

    // -------- phase 1: async-stage RGB halo tiles into LDS --------
    // LDS cell (r,c) <-> image pixel (qy0 + r, qx0 - HALO_L + c)
    const int NHALO = G_H * G_COLS; // 1349
    for (int idx = tid; idx < NHALO; idx += 256) {
        const int r  = idx / G_COLS;
        const int c  = idx - r * G_COLS;
        const int gy = qy0 + r;
        const int gx = qx0 - HALO_L + c;
        if ((unsigned)gy < (unsigned)H_IMG && (unsigned)gx < (unsigned)W_IMG) {
            const size_t off = (size_t)gy * W_IMG + gx;
            async_load_f32(pbase + off,              &s_raw[0][r][c]);
            async_load_f32(pbase + plane + off,      &s_raw[1][r][c]);
            async_load_f32(pbase + 2 * plane + off,  &s_raw[2][r][c]);
            async_load_f32(tbase + off,              &s_raw[3][r][c]);
            async_load_f32(tbase + plane + off,      &s_raw[4][r][c]);
            async_load_f32(tbase + 2 * plane + off,  &s_raw[5][r][c]);
        }
    }
    wait_async0();
    __syncthreads();

    // -------- phase 2: gray conversion; zero outside image (== jnp.pad) --------
    for (int idx = tid; idx < NHALO; idx += 256) {
        const int r  = idx / G_COLS;
        const int c  = idx - r * G_COLS;
        const int gy = qy0 + r;
        const int gx = qx0 - HALO_L + c;
        float gp = 0.0f, gt = 0.0f;
        if ((unsigned)gy < (unsigned)H_IMG && (unsigned)gx < (unsigned)W_IMG) {
            gp = fmaf(0.2989f, s_raw[0][r][c],
                 fmaf(0.587f,  s_raw[1][r][c], 0.114f * s_raw[2][r][c]));
            gt = fmaf(0.2989f, s_raw[3][r][c],
                 fmaf(0.587f,  s_raw[4][r][c], 0.114f * s_raw[5][r][c]));
        }
        s_gp[r][c] = gp;
        s_gt[r][c] = gt;
    }
    __syncthreads();

    // -------- phase 3: half-set census hamming, 4 horiz owners per thread ----
    // Sum over owners q and 24 offsets O+ of (m(q)+m(q+o)) * h(q,o).
    const int t_x = tid & 15;            // 16 groups of 4 pixels -> 64 cols
    const int t_y = tid >> 4;            // 16 rows
    const int c0  = 4 * t_x;             // 16B-aligned LDS col of window
    const int qy  = qy0 + t_y;           // owner row
    const int qx  = qx0 + 4 * t_x;       // first owner col

    // interior-mask predicates as 0/1 floats (mask has 1-px zero border)
    float ryf[4];
#pragma unroll
    for (int dy = 0; dy < 4; ++dy)
        ryf[dy] = (qy + dy >= 1 && qy + dy <= H_IMG - 2) ? 1.0f : 0.0f;
    float colw[12];
#pragma unroll
    for (int j = 0; j < 12; ++j) {
        const int x = qx - 4 + j;        // same indexing as loaded data
        colw[j] = (x >= 1 && x <= W_IMG - 2) ? 1.0f : 0.0f;
    }
    float mq[4];
#pragma unroll
    for (int i = 0; i < 4; ++i) mq[i] = ryf[0] * colw[4 + i];  // m(q)

    // centers = dy=0 row, loaded once and reused for the dy=0 offsets
    const float4* cp4 = (const float4*)(&s_gp[t_y][c0]);
    const float4* ct4 = (const float4*)(&s_gt[t_y][c0]);
    const float4 a0 = cp4[0], a1 = cp4[1], a2 = cp4[2];
    const float4 b0 = ct4[0], b1 = ct4[1], b2 = ct4[2];
    const float pv0[12] = {a0.x,a0.y,a0.z,a0.w, a1.x,a1.y,a1.z,a1.w, a2.x,a2.y,a2.z,a2.w};
    const float tv0[12] = {b0.x,b0.y,b0.z,b0.w, b1.x,b1.y,b1.z,b1.w, b2.x,b2.y,b2.z,b2.w};
    const float cp[4] = {pv0[4], pv0[5], pv0[6], pv0[7]};
    const float ct[4] = {tv0[4], tv0[5], tv0[6], tv0[7]};
    float acc[4] = {0.f, 0.f, 0.f, 0.f};

    // dy = 0, dx in {1,2,3}
#pragma unroll
    for (int dxi = 1; dxi <= 3; ++dxi) {
#pragma unroll
        for (int i = 0; i < 4; ++i) {
            const int j = dxi + 4 + i;                 // 5..10
            const float w  = fmaf(ryf[0], colw[j], mq[i]);
            const float tp = pv0[j] - cp[i];
            const float av = tp * __builtin_amdgcn_rsqf(fmaf(tp, tp, 0.81f));
            const float tq = tv0[j] - ct[i];
            const float bv = tq * __builtin_amdgcn_rsqf(fmaf(tq, tq, 0.81f));
            const float df = av - bv;
            const float rc = __builtin_amdgcn_rcpf(fmaf(df, df, 0.1f));
            acc[i] = fmaf(w * (df * df), rc, acc[i]);
        }
    }

    // dy = 1..3, dx in [-3,3]
#pragma unroll 1
    for (int dy = 1; dy < 4; ++dy) {
        const int rr = t_y + dy;
        const float4* rp4 = (const float4*)(&s_gp[rr][c0]);
        const float4* rt4 = (const float4*)(&s_gt[rr][c0]);
        const float4 p0 = rp4[0], p1 = rp4[1], p2 = rp4[2];
        const float4 q0 = rt4[0], q1 = rt4[1], q2 = rt4[2];
        const float pv[12] = {p0.x,p0.y,p0.z,p0.w, p1.x,p1.y,p1.z,p1.w, p2.x,p2.y,p2.z,p2.w};
        const float tv[12] = {q0.x,q0.y,q0.z,q0.w, q1.x,q1.y,q1.z,q1.w, q2.x,q2.y,q2.z,q2.w};
        const float ry = ryf[dy];
#pragma unroll
        for (int dx = 0; dx < 7; ++dx) {
#pragma unroll
            for (int i = 0; i < 4; ++i) {
                const int j = dx + 1 + i;              // dxi = dx-3 -> j = dxi+4+i, 1..10
                const float w  = fmaf(ry, colw[j], mq[i]);
                const float tp = pv[j] - cp[i];
                const float av = tp * __builtin_amdgcn_rsqf(fmaf(tp, tp, 0.81f));
                const float tq = tv[j] - ct[i];
                const float bv = tq * __builtin_amdgcn_rsqf(fmaf(tq, tq, 0.81f));
                const float df = av - bv;
                const float rc = __builtin_amdgcn_rcpf(fmaf(df, df, 0.1f));
                acc[i] = fmaf(w * (df * df), rc, acc[i]);
            }
        }
    }

    float tsum = (acc[0] + acc[1]) + (acc[2] + acc[3]);

    // -------- deterministic block reduction (wave32 shuffles) --------
#pragma unroll
    for (int off = 16; off > 0; off >>= 1)
        tsum += __shfl_down(tsum, off, 32);
    const int lane = tid & 31, wid = tid >> 5;
    if (lane == 0) s_red[wid] = tsum;
    __syncthreads();
    if (tid == 0) {
        float s = 0.0f;
#pragma unroll
        for (int w = 0; w < 8; ++w) s += s_red[w];
        partials[blockIdx.x + TILES_X * (blockIdx.y + TILES_Y * blockIdx.z)] = s;
    }
}

__global__ __launch_bounds__(256)
void census_reduce_kernel(const float* __restrict__ partials, float* __restrict__ out, int n)
{
    __shared__ float s_red[8];
    float s = 0.0f;
    for (int i = threadIdx.x; i < n; i += 256) s += partials[i];
#pragma unroll
    for (int off = 16; off > 0; off >>= 1)
        s += __shfl_down(s, off, 32);
    const int lane = threadIdx.x & 31, wid = threadIdx.x >> 5;
    if (lane == 0) s_red[wid] = s;
    __syncthreads();
    if (threadIdx.x == 0) {
        float t = 0.0f;
#pragma unroll
        for (int w = 0; w < 8; ++w) t += s_red[w];
        // mean over 49 census channels and over B*H*W pixels
        out[0] = t * (1.0f / (49.0f * (float)(BATCH * H_IMG * W_IMG)));
    }
}

extern "C" void kernel_launch(void* const* d_in, const int* in_sizes, int n_in,
                              void* d_out, int out_size, void* d_ws, size_t ws_size,
                              hipStream_t stream) {
    (void)in_sizes; (void)n_in; (void)out_size; (void)ws_size;
    const float* pred   = (const float*)d_in[0];
    const float* target = (const float*)d_in[1];
    float* partials     = (float*)d_ws;         // NBLOCKS floats, fully rewritten each call
    float* out          = (float*)d_out;

    dim3 grid(TILES_X, TILES_Y, BATCH);
    census_loss_kernel<<<grid, 256, 0, stream>>>(pred, target, partials);
    census_reduce_kernel<<<1, 256, 0, stream>>>(partials, out, NBLOCKS);
}